// GAT8Model_6124623364716
// MI455X (gfx1250) — compile-verified
//
#include <hip/hip_runtime.h>
#include <math.h>

// ---------------------------------------------------------------------------
// GATv2 x4 + conv head for MI455X (gfx1250, wave32).
//  - Node GEMMs (xl/xr) run on the matrix pipe: v_wmma_f32_16x16x4_f32 (fp32,
//    full precision, dual-output per wave to reuse the A fragment).
//  - Edge phase uses native f32 global atomics (add/max) — working set
//    (xl+xr+acc ~154MB) fits the 192MB L2, so gathers/scatters stay on-chip.
// Input pointer order assumes setup_inputs() dict insertion order, with the
// nested params dict flattened in insertion order:
//   0 features [N,36]  1 edge_index [2,E] (int)  2 edge_weights [E,1]
//   3..9   g1: Wl bl Wr br We att bias      (din=36, dout=64)
//   10..16 g2: (64,64)   17..23 g3: (64,64)  24..30 g4: (64,128)
//   31 c1_w 32 c1_b 33 c2_w 34 c2_b 35 c3_w 36 c3_b 37 l1_W(46) 38 l1_b
// ---------------------------------------------------------------------------

typedef __attribute__((ext_vector_type(2))) float v2f;
typedef __attribute__((ext_vector_type(8))) float v8f;

#define NEG_SLOPE 0.2f

__device__ __forceinline__ void atomicMaxF(float* addr, float val) {
  // Monotone bit tricks: valid because mx[] is initialized to -inf.
  if (val >= 0.0f)
    atomicMax((int*)addr, __float_as_int(val));
  else
    atomicMin((unsigned int*)addr, __float_as_uint(val));
}

// ---------------- init: zero accumulators, -inf maxes ----------------------
__global__ __launch_bounds__(256) void gat_init(float* __restrict__ mx,
                                                float* __restrict__ denom,
                                                float* __restrict__ acc,
                                                int n_nodes, int total_acc) {
  int i = blockIdx.x * 256 + threadIdx.x;
  if (i < total_acc) acc[i] = 0.0f;
  if (i < n_nodes) {
    mx[i] = -__builtin_inff();
    denom[i] = 0.0f;
  }
}

// ---------------- dual GEMM: XL = X*Wl^T + bl, XR = X*Wr^T + br ------------
// One wave -> one 16x16 tile of both outputs. fp32 WMMA 16x16x4.
// A frag (16x4 f32): lane 0-15 row M, VGPR pair holds K = 2*(lane>>4)+{0,1}.
// B frag (4x16 f32): lane 0-15 col N, same K striping. B[k][c] = W[c][k], so
// both fragments are contiguous float2 loads from row-major X / W.
__global__ __launch_bounds__(256) void gat_dual_gemm(
    const float* __restrict__ X, int n_nodes, int cin, int cout,
    const float* __restrict__ Wl, const float* __restrict__ bl,
    const float* __restrict__ Wr, const float* __restrict__ br,
    float* __restrict__ XL, float* __restrict__ XR) {
  const int lane = threadIdx.x & 31;
  const int tiles_n = cout >> 4;
  const int tiles_m = n_nodes >> 4;   // N=100000 -> 6250 exact
  const int wid = blockIdx.x * 8 + (threadIdx.x >> 5);
  if (wid >= tiles_m * tiles_n) return;   // wave-uniform: EXEC stays all-1s
  const int tm = wid / tiles_n;
  const int tn = wid - tm * tiles_n;
  const int r15 = lane & 15;
  const int hi  = lane >> 4;

  const float* xrow  = X  + (size_t)(tm * 16 + r15) * cin;
  const float* wlrow = Wl + (size_t)(tn * 16 + r15) * cin;
  const float* wrrow = Wr + (size_t)(tn * 16 + r15) * cin;

  v8f accl = {};
  v8f accr = {};
  for (int k0 = 0; k0 < cin; k0 += 4) {
    const int k = k0 + 2 * hi;
    v2f a  = *(const v2f*)(xrow + k);
    v2f bL = *(const v2f*)(wlrow + k);
    v2f bR = *(const v2f*)(wrrow + k);
    accl = __builtin_amdgcn_wmma_f32_16x16x4_f32(false, a, false, bL,
                                                 (short)0, accl, false, false);
    accr = __builtin_amdgcn_wmma_f32_16x16x4_f32(false, a, false, bR,
                                                 (short)0, accr, false, false);
  }

  const int col = tn * 16 + r15;
  const float biasL = bl[col];
  const float biasR = br[col];
#pragma unroll
  for (int i = 0; i < 8; ++i) {
    const int row = tm * 16 + i + hi * 8;   // D: VGPR i -> M=i / M=i+8
    XL[(size_t)row * cout + col] = accl[i] + biasL;
    XR[(size_t)row * cout + col] = accr[i] + biasR;
  }
}

// ---------------- per-edge attention score + segment max -------------------
// One wave per edge; lanes stride channels (C in {64,128}).
__global__ __launch_bounds__(256) void gat_edge_score(
    const int* __restrict__ src, const int* __restrict__ dst,
    const float* __restrict__ ew, const float* __restrict__ We,
    const float* __restrict__ att, const float* __restrict__ XL,
    const float* __restrict__ XR, int n_edges, int C,
    float* __restrict__ score, float* __restrict__ mx) {
  const int e = blockIdx.x * 8 + (threadIdx.x >> 5);
  if (e >= n_edges) return;
  const int lane = threadIdx.x & 31;
  const int s = src[e];
  const int d = dst[e];
  const float w = ew[e];
  float part = 0.0f;
  for (int c = lane; c < C; c += 32) {
    float m = XL[(size_t)s * C + c] + XR[(size_t)d * C + c] + w * We[c];
    float lr = (m > 0.0f) ? m : NEG_SLOPE * m;
    part += lr * att[c];
  }
#pragma unroll
  for (int off = 16; off > 0; off >>= 1) part += __shfl_xor(part, off, 32);
  if (lane == 0) {
    score[e] = part;
    atomicMaxF(&mx[d], part);
  }
}

// ---------------- exp-weight, denom and unnormalized aggregation -----------
__global__ __launch_bounds__(256) void gat_edge_accum(
    const int* __restrict__ src, const int* __restrict__ dst,
    const float* __restrict__ score, const float* __restrict__ mx,
    const float* __restrict__ XL, int n_edges, int C,
    float* __restrict__ denom, float* __restrict__ acc) {
  const int e = blockIdx.x * 8 + (threadIdx.x >> 5);
  if (e >= n_edges) return;
  const int lane = threadIdx.x & 31;
  const int s = src[e];
  const int d = dst[e];
  const float w = __expf(score[e] - mx[d]);
  if (lane == 0) unsafeAtomicAdd(&denom[d], w);   // global_atomic_add_f32
  for (int c = lane; c < C; c += 32)
    unsafeAtomicAdd(&acc[(size_t)d * C + c], w * XL[(size_t)s * C + c]);
}

// ---------------- normalize + bias + ReLU (in place on acc) ----------------
__global__ __launch_bounds__(256) void gat_finalize(
    float* __restrict__ acc, const float* __restrict__ denom,
    const float* __restrict__ bias, int n_nodes, int C) {
  const int i = blockIdx.x * 256 + threadIdx.x;
  if (i >= n_nodes * C) return;
  const int n = i / C;
  const int c = i - n * C;
  float dn = denom[n];
  dn = (dn > 1e-16f) ? dn : 1e-16f;
  float v = acc[i] / dn + bias[c];
  acc[i] = (v > 0.0f) ? v : 0.0f;
}

// ---------------- column mean over nodes -----------------------------------
__global__ __launch_bounds__(128) void zero128(float* __restrict__ p) {
  p[threadIdx.x] = 0.0f;
}

__global__ __launch_bounds__(128) void mean_reduce(
    const float* __restrict__ h, int n_nodes, float* __restrict__ mean) {
  const int c = threadIdx.x;           // 128 channels
  const float inv = 1.0f / (float)n_nodes;
  float p = 0.0f;
  for (int n = blockIdx.x; n < n_nodes; n += gridDim.x)
    p += h[(size_t)n * 128 + c];
  unsafeAtomicAdd(&mean[c], p * inv);
}

// ---------------- tiny conv/linear head (single block) ---------------------
__global__ __launch_bounds__(128) void head_kernel(
    const float* __restrict__ mean,
    const float* __restrict__ c1w, const float* __restrict__ c1b,
    const float* __restrict__ c2w, const float* __restrict__ c2b,
    const float* __restrict__ c3w, const float* __restrict__ c3b,
    const float* __restrict__ l1W, const float* __restrict__ l1b,
    float* __restrict__ out) {
  __shared__ float b0[128], b1[116], b2[104], b3[46];
  const int t = threadIdx.x;
  b0[t] = mean[t];
  __syncthreads();
  if (t < 116) {                         // conv1: K=5, dil=3 -> len 116
    float v = c1b[0];
#pragma unroll
    for (int k = 0; k < 5; ++k) v += b0[t + 3 * k] * c1w[k];
    b1[t] = (v > 0.0f) ? v : 0.0f;
  }
  __syncthreads();
  if (t < 104) {                         // conv2: K=5, dil=3 -> len 104
    float v = c2b[0];
#pragma unroll
    for (int k = 0; k < 5; ++k) v += b1[t + 3 * k] * c2w[k];
    b2[t] = (v > 0.0f) ? v : 0.0f;
  }
  __syncthreads();
  if (t < 46) {                          // conv3: K=5, dil=3, stride=2 -> 46
    float v = c3b[0];
#pragma unroll
    for (int k = 0; k < 5; ++k) v += b2[2 * t + 3 * k] * c3w[k];
    b3[t] = (v > 0.0f) ? v : 0.0f;
  }
  __syncthreads();
  if (t == 0) {                          // final linear -> scalar
    float v = l1b[0];
    for (int i = 0; i < 46; ++i) v += b3[i] * l1W[i];
    out[0] = v;
  }
}

// ---------------------------------------------------------------------------
extern "C" void kernel_launch(void* const* d_in, const int* in_sizes, int n_in,
                              void* d_out, int out_size, void* d_ws,
                              size_t ws_size, hipStream_t stream) {
  const int N = in_sizes[0] / 36;        // 100000
  const int E = in_sizes[2];             // 1600000

  const float* features = (const float*)d_in[0];
  const int*   ei       = (const int*)d_in[1];
  const float* ew       = (const float*)d_in[2];
  const int* src = ei;
  const int* dst = ei + E;

  struct GatP {
    const float *Wl, *bl, *Wr, *br, *We, *att, *bias;
    int din, dout;
  };
  auto f = [&](int i) { return (const float*)d_in[i]; };
  GatP g[4];
  const int dins[4]  = {36, 64, 64, 64};
  const int douts[4] = {64, 64, 64, 128};
  for (int L = 0; L < 4; ++L) {
    int b = 3 + L * 7;
    g[L] = {f(b + 0), f(b + 1), f(b + 2), f(b + 3),
            f(b + 4), f(b + 5), f(b + 6), dins[L], douts[L]};
  }
  const float* c1w = f(31); const float* c1b = f(32);
  const float* c2w = f(33); const float* c2b = f(34);
  const float* c3w = f(35); const float* c3b = f(36);
  const float* l1W = f(37); const float* l1b = f(38);

  // --- workspace layout (floats, 256B aligned chunks) ---
  char* ws = (char*)d_ws;
  size_t off = 0;
  auto alloc = [&](size_t nfloat) {
    float* p = (float*)(ws + off);
    off += ((nfloat * sizeof(float)) + 255) & ~(size_t)255;
    return p;
  };
  float* hA    = alloc((size_t)N * 128);
  float* hB    = alloc((size_t)N * 128);
  float* xl    = alloc((size_t)N * 128);
  float* xr    = alloc((size_t)N * 128);
  float* score = alloc((size_t)E);
  float* mx    = alloc((size_t)N);
  float* denom = alloc((size_t)N);
  float* mean  = alloc(128);
  (void)ws_size; (void)n_in; (void)out_size;

  const float* x_cur = features;
  float* bufs[2] = {hA, hB};

  const int edge_blocks = (E + 7) / 8;   // 1 wave per edge, 8 waves/block

  for (int L = 0; L < 4; ++L) {
    const int din = g[L].din, dout = g[L].dout;
    float* acc = bufs[L & 1];            // layer output, used as accumulator

    // init: acc=0, mx=-inf, denom=0
    {
      int total = N * dout;
      gat_init<<<(total + 255) / 256, 256, 0, stream>>>(mx, denom, acc, N,
                                                        total);
    }
    // dual WMMA GEMM -> xl, xr
    {
      int waves = (N >> 4) * (dout >> 4);
      gat_dual_gemm<<<(waves + 7) / 8, 256, 0, stream>>>(
          x_cur, N, din, dout, g[L].Wl, g[L].bl, g[L].Wr, g[L].br, xl, xr);
    }
    // edge scores + segment max
    gat_edge_score<<<edge_blocks, 256, 0, stream>>>(
        src, dst, ew, g[L].We, g[L].att, xl, xr, E, dout, score, mx);
    // exp weights, denom, unnormalized aggregation
    gat_edge_accum<<<edge_blocks, 256, 0, stream>>>(src, dst, score, mx, xl,
                                                    E, dout, denom, acc);
    // normalize + bias + ReLU in place
    gat_finalize<<<(N * dout + 255) / 256, 256, 0, stream>>>(acc, denom,
                                                             g[L].bias, N,
                                                             dout);
    x_cur = acc;
  }

  // mean over nodes (layer-4 output is in bufs[1], C=128)
  zero128<<<1, 128, 0, stream>>>(mean);
  mean_reduce<<<256, 128, 0, stream>>>(bufs[1], N, mean);

  // conv head + final linear -> scalar
  head_kernel<<<1, 128, 0, stream>>>(mean, c1w, c1b, c2w, c2b, c3w, c3b, l1W,
                                     l1b, (float*)d_out);
}